// AttentionHead_84018150244852
// MI455X (gfx1250) — compile-verified
//
#include <hip/hip_runtime.h>

// Problem constants (match reference)
#define B_  8
#define S_  2048
#define D_  768
#define H_  12
#define HP  16            // padded head dim for WMMA tiles
#define NROWS (B_*S_)     // 16384 flattened (b,s) rows

typedef float v2f __attribute__((ext_vector_type(2)));
typedef float v8f __attribute__((ext_vector_type(8)));

__device__ __forceinline__ v8f wmma_f32(v2f a, v2f b, v8f c) {
  // D = A(16x4 f32) * B(4x16 f32) + C(16x16 f32), full fp32 precision
  return __builtin_amdgcn_wmma_f32_16x16x4_f32(false, a, false, b, (short)0, c,
                                               false, false);
}

// ---------------------------------------------------------------------------
// Kernel 0: transpose+pad weights (D,12) -> (16,D) so WMMA B operands are
// contiguous b64 loads. Order in scratch: [K, Q, V].
// ---------------------------------------------------------------------------
__global__ void pad_weights_kernel(const float* __restrict__ wk,
                                   const float* __restrict__ wq,
                                   const float* __restrict__ wv,
                                   float* __restrict__ wt) {
  int idx = blockIdx.x * blockDim.x + threadIdx.x;
  const int total = 3 * HP * D_;
  if (idx >= total) return;
  int w   = idx / (HP * D_);
  int rem = idx - w * (HP * D_);
  int n   = rem / D_;          // padded head column 0..15
  int d   = rem - n * D_;      // feature row 0..767
  const float* src = (w == 0) ? wk : (w == 1) ? wq : wv;
  wt[idx] = (n < H_) ? src[d * H_ + n] : 0.0f;
}

// ---------------------------------------------------------------------------
// Kernel 1: fused K/Q/V projection. One wave per 16 rows of x.
// C(16 rows x 16 heads) accumulated over K=768 in steps of 4; the A operand
// (x tile) is shared by the three WMMAs so x is read exactly once.
// ---------------------------------------------------------------------------
__global__ void qkv_proj_kernel(const float* __restrict__ x,
                                const float* __restrict__ wt,
                                float* __restrict__ kbuf,
                                float* __restrict__ qbuf,
                                float* __restrict__ vbuf) {
  const int l    = threadIdx.x;
  const int nm   = l & 15;          // A: row m / B: col n
  const int half = l >> 4;          // K-subchunk selector
  const int kb2  = half * 2;
  const int row0 = blockIdx.x * 16;

  const float* xrow = x  + (size_t)(row0 + nm) * D_;   // A rows
  const float* wk   = wt + 0 * HP * D_ + nm * D_;      // B rows (transposed W)
  const float* wq   = wt + 1 * HP * D_ + nm * D_;
  const float* wv   = wt + 2 * HP * D_ + nm * D_;

  v8f ck = 0.0f, cq = 0.0f, cv = 0.0f;

#pragma unroll 4
  for (int k0 = 0; k0 < D_; k0 += 4) {
    v2f a  = *(const v2f*)(xrow + k0 + kb2);   // A(m, k0+kb2 .. +1)
    v2f bk = *(const v2f*)(wk   + k0 + kb2);   // B(k0+kb2.., n)
    v2f bq = *(const v2f*)(wq   + k0 + kb2);
    v2f bv = *(const v2f*)(wv   + k0 + kb2);
    ck = wmma_f32(a, bk, ck);
    cq = wmma_f32(a, bq, cq);
    cv = wmma_f32(a, bv, cv);
  }

  if (nm < H_) {   // only 12 valid head columns
#pragma unroll
    for (int r = 0; r < 8; ++r) {
      int m = r + 8 * half;                     // C row per ISA layout
      size_t o = (size_t)(row0 + m) * H_ + nm;
      kbuf[o] = ck[r];
      qbuf[o] = cq[r];
      vbuf[o] = cv[r];
    }
  }
}

// ---------------------------------------------------------------------------
// Kernel 2: flash attention, one wave per (batch, 16-query tile).
//   S^T = K_tile(16x12) * q^T(12x16)      -> scores live per-query per-lane
//   online softmax with key mask          -> p[], running m/l, acc rescale
//   O^T += V^T(16hx16k) * P^T(16kx16q)    -> 4 chained WMMAs
// ---------------------------------------------------------------------------
__global__ void flash_attn_kernel(const float* __restrict__ qbuf,
                                  const float* __restrict__ kbuf,
                                  const float* __restrict__ vbuf,
                                  const int*   __restrict__ mask,
                                  float*       __restrict__ out) {
  const int l    = threadIdx.x;
  const int nm   = l & 15;       // query column n (C layout) / matrix row m (A)
  const int half = l >> 4;
  const int kb2  = half * 2;
  const int b    = blockIdx.x / (S_ / 16);
  const int q0   = (blockIdx.x % (S_ / 16)) * 16;
  const float scale = 0.28867513459481287f;   // 1/sqrt(12)
  const size_t base = (size_t)b * S_;

  // B operand for scores: q^T, loaded once. B(k=h, n=query).
  const float* qrow = qbuf + (base + q0 + nm) * H_;
  v2f bq0 = *(const v2f*)(qrow + 0 + kb2);
  v2f bq1 = *(const v2f*)(qrow + 4 + kb2);
  v2f bq2 = *(const v2f*)(qrow + 8 + kb2);

  float mrow = -3.0e38f;   // running max (per query, replicated in both halves)
  float lsum = 0.0f;       // running denominator
  v8f acc = 0.0f;          // O^T accumulator (rows h = r + 8*half, col query nm)

  for (int t0 = 0; t0 < S_; t0 += 16) {
    // --- scores^T = K_tile * q^T  (A rows = keys, K dim = h, exactly 12) ---
    const float* krow = kbuf + (base + t0 + nm) * H_;
    v2f ak0 = *(const v2f*)(krow + 0 + kb2);
    v2f ak1 = *(const v2f*)(krow + 4 + kb2);
    v2f ak2 = *(const v2f*)(krow + 8 + kb2);
    v8f c = 0.0f;
    c = wmma_f32(ak0, bq0, c);
    c = wmma_f32(ak1, bq1, c);
    c = wmma_f32(ak2, bq2, c);
    // c[r] = score for key (t0 + r + 8*half), query nm

    // --- key mask: 8 consecutive keys starting at t0 + 8*half ---
    const int* mp = mask + base + t0 + 8 * half;
    int4 mv0 = *(const int4*)(mp);
    int4 mv1 = *(const int4*)(mp + 4);

    float s0 = mv0.x ? c[0] * scale : -1.0e30f;
    float s1 = mv0.y ? c[1] * scale : -1.0e30f;
    float s2 = mv0.z ? c[2] * scale : -1.0e30f;
    float s3 = mv0.w ? c[3] * scale : -1.0e30f;
    float s4 = mv1.x ? c[4] * scale : -1.0e30f;
    float s5 = mv1.y ? c[5] * scale : -1.0e30f;
    float s6 = mv1.z ? c[6] * scale : -1.0e30f;
    float s7 = mv1.w ? c[7] * scale : -1.0e30f;

    // --- online softmax: tile max (8 local + partner half via xor-16) ---
    float tmax = fmaxf(fmaxf(fmaxf(s0, s1), fmaxf(s2, s3)),
                       fmaxf(fmaxf(s4, s5), fmaxf(s6, s7)));
    tmax = fmaxf(tmax, __shfl_xor(tmax, 16, 32));
    float mnew = fmaxf(mrow, tmax);
    float corr = __expf(mrow - mnew);

    float p0 = __expf(s0 - mnew), p1 = __expf(s1 - mnew);
    float p2 = __expf(s2 - mnew), p3 = __expf(s3 - mnew);
    float p4 = __expf(s4 - mnew), p5 = __expf(s5 - mnew);
    float p6 = __expf(s6 - mnew), p7 = __expf(s7 - mnew);
    float psum = ((p0 + p1) + (p2 + p3)) + ((p4 + p5) + (p6 + p7));
    psum += __shfl_xor(psum, 16, 32);
    lsum = lsum * corr + psum;
    mrow = mnew;
#pragma unroll
    for (int r = 0; r < 8; ++r) acc[r] *= corr;

    // --- exchange the 4 p-values the partner half needs for B(P^T) ---
    float e0 = __shfl_xor(half ? p0 : p2, 16, 32);  // h0 gets po0 ; h1 gets po2
    float e1 = __shfl_xor(half ? p1 : p3, 16, 32);
    float e2 = __shfl_xor(half ? p4 : p6, 16, 32);  // h0 gets po4 ; h1 gets po6
    float e3 = __shfl_xor(half ? p5 : p7, 16, 32);

    // --- A = V^T chunks: A(m=h, k=key). 2 strided b32 loads per chunk. ---
    const float* vb0 = vbuf + (base + t0) * H_ + nm;
    v2f av0, av1, av2, av3;
    av0.x = vb0[(0  + kb2) * H_];  av0.y = vb0[(1  + kb2) * H_];
    av1.x = vb0[(4  + kb2) * H_];  av1.y = vb0[(5  + kb2) * H_];
    av2.x = vb0[(8  + kb2) * H_];  av2.y = vb0[(9  + kb2) * H_];
    av3.x = vb0[(12 + kb2) * H_];  av3.y = vb0[(13 + kb2) * H_];

    // --- B = P^T chunks: B(k=key, n=query). chunk j needs keys 4j+2*half. ---
    v2f bp0, bp1, bp2, bp3;
    bp0.x = half ? e0 : p0;  bp0.y = half ? e1 : p1;  // keys {0,1}/{2,3}
    bp1.x = half ? e2 : p4;  bp1.y = half ? e3 : p5;  // keys {4,5}/{6,7}
    bp2.x = half ? p2 : e0;  bp2.y = half ? p3 : e1;  // keys {8,9}/{10,11}
    bp3.x = half ? p6 : e2;  bp3.y = half ? p7 : e3;  // keys {12,13}/{14,15}

    acc = wmma_f32(av0, bp0, acc);
    acc = wmma_f32(av1, bp1, acc);
    acc = wmma_f32(av2, bp2, acc);
    acc = wmma_f32(av3, bp3, acc);
  }

  const float inv = 1.0f / lsum;
#pragma unroll
  for (int r = 0; r < 8; ++r) {
    int h = r + 8 * half;            // O^T row = head dim
    if (h < H_)
      out[(base + q0 + nm) * H_ + h] = acc[r] * inv;
  }
}

// ---------------------------------------------------------------------------
// Launch: ws layout (floats):
//   [0, 3*16*768)                       padded-transposed weights K,Q,V
//   then K, Q, V buffers of NROWS*12 each, + 64 floats slack (the V^T loads
//   for padded head rows 12..15 overshoot the last row by up to 4 floats).
// Total ~2.51 MB of scratch.
// ---------------------------------------------------------------------------
extern "C" void kernel_launch(void* const* d_in, const int* in_sizes, int n_in,
                              void* d_out, int out_size, void* d_ws, size_t ws_size,
                              hipStream_t stream) {
  const float* x    = (const float*)d_in[0];
  const int*   mask = (const int*)  d_in[1];
  const float* wk   = (const float*)d_in[2];
  const float* wq   = (const float*)d_in[3];
  const float* wv   = (const float*)d_in[4];
  float* out = (float*)d_out;

  float* wt   = (float*)d_ws;
  float* kbuf = wt   + 3 * HP * D_;
  float* qbuf = kbuf + (size_t)NROWS * H_;
  float* vbuf = qbuf + (size_t)NROWS * H_;

  const int padN = 3 * HP * D_;
  pad_weights_kernel<<<(padN + 255) / 256, 256, 0, stream>>>(wk, wq, wv, wt);
  qkv_proj_kernel<<<NROWS / 16, 32, 0, stream>>>(x, wt, kbuf, qbuf, vbuf);
  flash_attn_kernel<<<B_ * (S_ / 16), 32, 0, stream>>>(qbuf, kbuf, vbuf, mask, out);
}